// OLLoss10_29918742184752
// MI455X (gfx1250) — compile-verified
//
#include <hip/hip_runtime.h>
#include <hip/hip_bf16.h>

// OLLoss: loss = mean_b( sum_c( -log(1 - softmax(logits)_bc) * dist[label_b, c] ) )
// B = 2097152, C = 10.  Memory-bound: ~88 MB in -> ~3.8 us floor @ 23.3 TB/s.
// CDNA5 path used: global_load_async_to_lds_b128 (ASYNCcnt) to stage contiguous
// logits tiles into LDS with perfectly coalesced 128b lane-contiguous requests.

#define C_CLASSES 10
#define TPB 256
#define SAMPLES_PER_THREAD 2
#define TILE_SAMPLES (TPB * SAMPLES_PER_THREAD)          // 512 samples / block
#define TILE_FLOATS (TILE_SAMPLES * C_CLASSES)           // 5120 floats
#define TILE_VEC4 (TILE_FLOATS / 4)                      // 1280 float4 (20 KB)
#define VEC4_PER_THREAD (TILE_VEC4 / TPB)                // 5

#if defined(__has_builtin)
#if __has_builtin(__builtin_amdgcn_global_load_async_to_lds_b128)
#define HAVE_ASYNC_LDS 1
#endif
#if __has_builtin(__builtin_amdgcn_s_wait_asynccnt)
#define HAVE_WAIT_ASYNC 1
#endif
#endif

// Builtin prototype (from hipcc diagnostic): params are pointers to
// 16-byte int vectors, generic address space.
typedef int v4i_t __attribute__((vector_size(16)));

__device__ __forceinline__ void async_copy16(const float4* gsrc, float4* ldst) {
#if defined(HAVE_ASYNC_LDS)
  __builtin_amdgcn_global_load_async_to_lds_b128(
      (v4i_t*)const_cast<float4*>(gsrc), (v4i_t*)ldst, 0, 0);
#else
  *ldst = *gsrc;   // fallback: normal load + ds_store (still b128 both ways)
#endif
}

__device__ __forceinline__ void async_wait0() {
#if defined(HAVE_ASYNC_LDS)
#if defined(HAVE_WAIT_ASYNC)
  __builtin_amdgcn_s_wait_asynccnt(0);
#else
  asm volatile("s_wait_asynccnt 0" ::: "memory");
#endif
#endif
  asm volatile("" ::: "memory");
}

// Per-row loss: softmax in registers + hardware exp/log (v_exp_f32 / v_log_f32).
__device__ __forceinline__ float row_loss(const float x[C_CLASSES],
                                          const float* __restrict__ drow) {
  float m = x[0];
#pragma unroll
  for (int c = 1; c < C_CLASSES; ++c) m = fmaxf(m, x[c]);
  float e[C_CLASSES];
  float S = 0.0f;
#pragma unroll
  for (int c = 0; c < C_CLASSES; ++c) {
    e[c] = __expf(x[c] - m);
    S += e[c];
  }
  const float rS = __frcp_rn(S);
  float acc = 0.0f;
#pragma unroll
  for (int c = 0; c < C_CLASSES; ++c) {
    const float p = e[c] * rS;
    const float q = fmaxf(1.0f - p, 1e-12f);   // guard -inf*0 at d==0
    acc = fmaf(-__logf(q), drow[c], acc);
  }
  return acc;
}

__global__ void __launch_bounds__(TPB) ol_loss_main(
    const float* __restrict__ logits,   // [B, 10]
    const int* __restrict__ labels,     // [B]  (harness ABI: integer -> int32)
    const float* __restrict__ dist,     // [10, 10]
    float* __restrict__ partials,       // [gridDim.x]
    int nB) {
  __shared__ float4 tile4[TILE_VEC4];      // 20 KB staged logits tile
  __shared__ float sdist[C_CLASSES * C_CLASSES];
  __shared__ float wsum[TPB / 32];

  const int t = threadIdx.x;
  const int tile0 = blockIdx.x * TILE_SAMPLES;

  if (t < C_CLASSES * C_CLASSES) sdist[t] = dist[t];

  // Stage tile: lane-contiguous b128 async loads (fully coalesced globally).
  const float4* g4 = (const float4*)logits + (size_t)blockIdx.x * TILE_VEC4;
  const long totalF = (long)nB * C_CLASSES;
#pragma unroll
  for (int k = 0; k < VEC4_PER_THREAD; ++k) {
    const int idx = k * TPB + t;
    const long gFloat = ((long)blockIdx.x * TILE_VEC4 + idx) * 4;
    if (gFloat < totalF) async_copy16(g4 + idx, tile4 + idx);
  }
  async_wait0();
  __syncthreads();

  const int s0 = tile0 + SAMPLES_PER_THREAD * t;
  int2 lab = make_int2(0, 0);
  if (s0 + 1 < nB)      lab = *(const int2*)(labels + s0);
  else if (s0 < nB)     lab.x = labels[s0];

  const float* tf = (const float*)tile4;
  float acc = 0.0f;
#pragma unroll
  for (int r = 0; r < SAMPLES_PER_THREAD; ++r) {
    const int s = s0 + r;
    if (s < nB) {
      float x[C_CLASSES];
      const float* row = tf + (SAMPLES_PER_THREAD * t + r) * C_CLASSES;
#pragma unroll
      for (int c = 0; c < C_CLASSES; ++c) x[c] = row[c];
      const int lb = (r == 0) ? lab.x : lab.y;
      acc += row_loss(x, &sdist[lb * C_CLASSES]);
    }
  }

  // Deterministic block reduction: wave32 shfl_xor tree + LDS across 8 waves.
#pragma unroll
  for (int m = 16; m > 0; m >>= 1) acc += __shfl_xor(acc, m, 32);
  if ((t & 31) == 0) wsum[t >> 5] = acc;
  __syncthreads();
  if (t == 0) {
    float s = 0.0f;
#pragma unroll
    for (int i = 0; i < TPB / 32; ++i) s += wsum[i];
    partials[blockIdx.x] = s;
  }
}

__global__ void __launch_bounds__(TPB) ol_loss_reduce(
    const float* __restrict__ partials, float* __restrict__ out,
    int n, float invB) {
  __shared__ float wsum[TPB / 32];
  float s = 0.0f;
  for (int i = threadIdx.x; i < n; i += TPB) s += partials[i];
#pragma unroll
  for (int m = 16; m > 0; m >>= 1) s += __shfl_xor(s, m, 32);
  if ((threadIdx.x & 31) == 0) wsum[threadIdx.x >> 5] = s;
  __syncthreads();
  if (threadIdx.x == 0) {
    float tot = 0.0f;
#pragma unroll
    for (int i = 0; i < TPB / 32; ++i) tot += wsum[i];
    out[0] = tot * invB;
  }
}

extern "C" void kernel_launch(void* const* d_in, const int* in_sizes, int n_in,
                              void* d_out, int out_size, void* d_ws, size_t ws_size,
                              hipStream_t stream) {
  const float* logits = (const float*)d_in[0];
  const int* labels   = (const int*)d_in[1];
  const float* dist   = (const float*)d_in[2];
  float* out          = (float*)d_out;
  float* partials     = (float*)d_ws;

  const int nB = in_sizes[1];                      // B from labels' element count
  const int blocks = (nB + TILE_SAMPLES - 1) / TILE_SAMPLES;   // 4096 for B=2^21

  ol_loss_main<<<blocks, TPB, 0, stream>>>(logits, labels, dist, partials, nB);
  ol_loss_reduce<<<1, TPB, 0, stream>>>(partials, out, blocks, 1.0f / (float)nB);
}